// DNPUConv2d_86612310491664
// MI455X (gfx1250) — compile-verified
//
#include <hip/hip_runtime.h>

// ---- problem constants (match reference) ----
#define IN_CH   8
#define OUT_CH  16
#define DEV     3
#define N_IN    3
#define N_CTRL  4
#define HID     90
#define HP      96      // HID padded to multiple of 16
#define BATCH   16
#define HW      16
#define LPIX    256     // 16x16 output positions
#define KT      24      // HP/4 K-fragments for 16x16x4 wmma
#define NT      6       // HP/16 N-tiles

typedef __attribute__((ext_vector_type(2))) float v2f;
typedef __attribute__((ext_vector_type(8))) float v8f;

// ---- workspace layout (float offsets) ----
#define WS_W2P    0
#define WS_CBIAS  (WS_W2P   + HP*HP)                    // [O][I][D][HP]
#define WS_W1U    (WS_CBIAS + OUT_CH*IN_CH*DEV*HP)      // [D][3][HP]
#define WS_B2P    (WS_W1U   + DEV*3*HP)
#define WS_W3P    (WS_B2P   + HP)
#define WS_TOTAL  (WS_W3P   + HP)

// Find position of raw-electrode k within idx_d = [input_indices[d], control_indices[d]]
__device__ __forceinline__ int elec_pos(const int* in_idx, const int* ct_idx, int d, int raw) {
    int pos = 0;
    for (int e = 0; e < N_IN;  ++e) if (in_idx[d*N_IN  + e] == raw) pos = e;
    for (int e = 0; e < N_CTRL;++e) if (ct_idx[d*N_CTRL + e] == raw) pos = N_IN + e;
    return pos;
}

// Precompute: permuted/padded weights + folded control bias into ws.
__global__ void dnpu_pre(const float* __restrict__ ctrl, const float* __restrict__ W1,
                         const float* __restrict__ b1,   const float* __restrict__ W2,
                         const float* __restrict__ b2,   const float* __restrict__ W3,
                         const int* __restrict__ in_idx, const int* __restrict__ ct_idx,
                         float* __restrict__ ws) {
    int tid = blockIdx.x * blockDim.x + threadIdx.x;
    if (tid < WS_CBIAS) {                                   // zero-padded W2 [HP][HP]
        int r = tid / HP, c = tid % HP;
        ws[WS_W2P + tid] = (r < HID && c < HID) ? W2[r*HID + c] : 0.f;
    } else if (tid < WS_W1U) {                              // Cbias[o][i][d][h]
        int t = tid - WS_CBIAS;
        int h = t % HP, d = (t / HP) % DEV;
        int i = (t / (HP*DEV)) % IN_CH, o = t / (HP*DEV*IN_CH);
        float v = 0.f;
        if (h < HID) {
            v = b1[h];
            for (int j = 0; j < N_CTRL; ++j) {
                int pos = elec_pos(in_idx, ct_idx, d, N_IN + j);
                v += ctrl[((o*IN_CH + i)*DEV + d)*N_CTRL + j] * W1[pos*HID + h];
            }
        }
        ws[tid] = v;
    } else if (tid < WS_B2P) {                              // W1u[d][j][h] (u-electrode rows)
        int t = tid - WS_W1U;
        int h = t % HP, j = (t / HP) % 3, d = t / (HP*3);
        float v = 0.f;
        if (h < HID) {
            int pos = elec_pos(in_idx, ct_idx, d, j);
            v = W1[pos*HID + h];
        }
        ws[tid] = v;
    } else if (tid < WS_W3P) {                              // b2 padded
        int h = tid - WS_B2P;
        ws[tid] = (h < HID) ? b2[h] : 0.f;
    } else if (tid < WS_TOTAL) {                            // W3 padded (shape [90,1])
        int h = tid - WS_W3P;
        ws[tid] = (h < HID) ? W3[h] : 0.f;
    }
}

__device__ __forceinline__ float elu1(float x) {
    return x > 0.f ? x : (__expf(x) - 1.f);
}

// Main fused kernel: one wave = one 16-row tile (b, o, oh=lt), loops 24 (i,d) combos.
__global__ void __launch_bounds__(256)
__attribute__((amdgpu_waves_per_eu(4)))
dnpu_main(const float* __restrict__ x, const float* __restrict__ b3,
          const float* __restrict__ ws, float* __restrict__ out) {
    __shared__ float s_w2[HP*HP];            // 36 KB padded W2
    __shared__ float s_cb[IN_CH*DEV*HP];     // Cbias slice for this o
    __shared__ float s_w1u[DEV*3*HP];
    __shared__ float s_b2[HP];
    __shared__ float s_w3[HP];
    __shared__ float s_x[IN_CH*HW*HW];       // x[b] slice, 8 KB

    const int t0 = blockIdx.x * 8;                 // tile id of wave 0
    const int b  = t0 / (OUT_CH * 16);             // uniform across block
    const int o  = (t0 / 16) % OUT_CH;             // uniform across block
    const int tid = threadIdx.x;

    for (int j = tid; j < HP*HP; j += 256)          s_w2[j]  = ws[WS_W2P + j];
    for (int j = tid; j < IN_CH*DEV*HP; j += 256)   s_cb[j]  = ws[WS_CBIAS + o*IN_CH*DEV*HP + j];
    for (int j = tid; j < DEV*3*HP; j += 256)       s_w1u[j] = ws[WS_W1U + j];
    if (tid < HP) { s_b2[tid] = ws[WS_B2P + tid];   s_w3[tid] = ws[WS_W3P + tid]; }
    for (int j = tid; j < IN_CH*HW*HW; j += 256)    s_x[j]   = x[b*IN_CH*HW*HW + j];
    __syncthreads();

    const int wave = tid >> 5;
    const int lane = tid & 31;
    const int lt   = (t0 + wave) & 15;     // oh index
    const int n    = lane & 15;            // ow index / N-column within tile
    const int half = lane >> 4;

    v2f   af[KT];                          // layer-1 activations, A-fragment layout
    float p[8];
    #pragma unroll
    for (int v = 0; v < 8; ++v) p[v] = 0.f;

    #pragma clang loop unroll(disable)
    for (int id = 0; id < IN_CH * DEV; ++id) {
        const int i = id / DEV, d = id % DEV;

        // u electrodes: x[b,i, lt+d-1, n-1..n+1], zero-padded
        float u0 = 0.f, u1 = 0.f, u2 = 0.f;
        const int yy = lt + d - 1;
        if (yy >= 0 && yy < HW) {
            const float* xr = &s_x[(i*HW + yy)*HW];
            if (n >= 1)     u0 = xr[n - 1];
            u1 = xr[n];
            if (n + 1 < HW) u2 = xr[n + 1];
        }

        // Opaque def of the Cbias offset: anchors this iteration's layer-1 LDS
        // loads so the scheduler cannot pipeline more than one iteration ahead.
        int cbo = (i*DEV + d)*HP;
        asm volatile("" : "+v"(cbo));

        // ---- layer 1 (VALU, written directly in WMMA A-fragment layout) ----
        const float* w1d = &s_w1u[d*3*HP];
        const float* cb  = &s_cb[cbo];
        #pragma unroll
        for (int kt = 0; kt < KT; ++kt) {
            #pragma unroll
            for (int v = 0; v < 2; ++v) {
                const int kk = 4*kt + v + 2*half;   // K index held by this lane/VGPR
                float h = cb[kk] + u0*w1d[kk] + u1*w1d[HP + kk] + u2*w1d[2*HP + kk];
                af[kt][v] = elu1(h);                 // elu(0)=0 keeps padding exact
            }
        }

        // ---- layer 2 via fp32 WMMA 16x16x4, fused layer-3 matvec ----
        #pragma unroll
        for (int nt = 0; nt < NT; ++nt) {
            // Opaque def of this N-tile's column offset: the W2 B-fragment LDS
            // reads below depend on it, so they stay in-loop (cheap conflict-free
            // LDS rereads) instead of being hoisted across the whole id loop.
            int col = 16*nt + n;
            asm volatile("" : "+v"(col));

            // Stage 1: batch all 24 B-fragments of this N-tile into registers.
            // All 24 paired ds_loads go in flight together -> one LDS latency
            // per N-tile instead of one per WMMA.
            v2f barr[KT];
            #pragma unroll
            for (int kt = 0; kt < KT; ++kt) {
                barr[kt][0] = s_w2[(4*kt + 0 + 2*half)*HP + col];
                barr[kt][1] = s_w2[(4*kt + 1 + 2*half)*HP + col];
            }

            // Stage 2: back-to-back 24-deep WMMA accumulation chain.
            const float b2n = s_b2[16*nt + n];
            v8f acc = { b2n, b2n, b2n, b2n, b2n, b2n, b2n, b2n };
            #pragma unroll
            for (int kt = 0; kt < KT; ++kt) {
                acc = __builtin_amdgcn_wmma_f32_16x16x4_f32(
                          false, af[kt], false, barr[kt], (short)0, acc, false, false);
            }

            const float w3n = s_w3[16*nt + n];
            #pragma unroll
            for (int v = 0; v < 8; ++v)
                p[v] += elu1(acc[v]) * w3n;          // layer 3 partial (per-lane column n)
        }
    }

    // reduce layer-3 partials over the 16 columns (lanes within each half)
    #pragma unroll
    for (int v = 0; v < 8; ++v) {
        float s = p[v];
        s += __shfl_xor(s, 1, 32);
        s += __shfl_xor(s, 2, 32);
        s += __shfl_xor(s, 4, 32);
        s += __shfl_xor(s, 8, 32);
        p[v] = s;
    }

    if (n == 0) {
        const float bias3 = b3[0] * (float)(IN_CH * DEV);   // b3 added per (i,d) row
        const int base = ((b*OUT_CH + o)*LPIX) + lt*16 + 8*half;
        #pragma unroll
        for (int v = 0; v < 8; ++v) out[base + v] = p[v] + bias3;
    }
}

extern "C" void kernel_launch(void* const* d_in, const int* in_sizes, int n_in,
                              void* d_out, int out_size, void* d_ws, size_t ws_size,
                              hipStream_t stream) {
    const float* x     = (const float*)d_in[0];
    const float* ctrl  = (const float*)d_in[1];
    const float* W1    = (const float*)d_in[2];
    const float* b1    = (const float*)d_in[3];
    const float* W2    = (const float*)d_in[4];
    const float* b2    = (const float*)d_in[5];
    const float* W3    = (const float*)d_in[6];
    const float* b3    = (const float*)d_in[7];
    const int*   iidx  = (const int*)d_in[8];
    const int*   cidx  = (const int*)d_in[9];
    float*       ws    = (float*)d_ws;
    float*       out   = (float*)d_out;

    dnpu_pre<<<(WS_TOTAL + 255) / 256, 256, 0, stream>>>(ctrl, W1, b1, W2, b2, W3,
                                                         iidx, cidx, ws);
    // tiles = B*O*(L/16) = 4096, 8 waves (tiles) per 256-thread block
    dnpu_main<<<512, 256, 0, stream>>>(x, b3, ws, out);
}